// NoisySyn_eld_4947802325767
// MI455X (gfx1250) — compile-verified
//
#include <hip/hip_runtime.h>
#include <cstdint>

// ---------------------------------------------------------------------------
// NoisySyn ELD noise synthesis, MI455X (gfx1250).
// Memory-bound elementwise kernel: 33.18M fp32 elements, ~398MB HBM traffic,
// ~17us floor at 23.3 TB/s. Strategy: b128 non-temporal loads/stores,
// async-to-LDS staging of the small n_row table, prefetch of the next tile,
// counter-based RNG (deterministic per element) for the Poisson sample.
// ---------------------------------------------------------------------------

typedef float f4 __attribute__((ext_vector_type(4)));
typedef int v4i __attribute__((ext_vector_type(4)));

#define GOLD 0x9E3779B97F4A7C15ull

#if __has_builtin(__builtin_amdgcn_global_load_async_to_lds_b128) && \
    __has_builtin(__builtin_amdgcn_s_wait_asynccnt)
#define USE_ASYNC_LDS 1
#endif

__device__ __forceinline__ uint64_t mix64(uint64_t z) {
  z = (z ^ (z >> 30)) * 0xBF58476D1CE4E5B9ull;
  z = (z ^ (z >> 27)) * 0x94D049BB133111EBull;
  return z ^ (z >> 31);
}

struct Rng {
  uint64_t s;
  __device__ __forceinline__ float next() {
    s += GOLD;
    uint32_t hi = (uint32_t)(mix64(s) >> 32);
    // uniform in (0,1), never exactly 0 or 1
    return fmaf((float)hi, 0x1p-32f, 0x1p-33f);
  }
};

// Poisson sampler: Knuth product method for small lambda, Hormann PTRS
// transformed rejection for lambda >= 10. Deterministic per seed.
__device__ __forceinline__ float poissonf(float lam, uint64_t seed) {
  if (!(lam > 0.0f)) return 0.0f;
  Rng r{seed};
  if (lam < 10.0f) {
    const float L = __expf(-lam);
    float p = 1.0f, k = -1.0f;
#pragma unroll 1
    do {
      k += 1.0f;
      p *= r.next();
    } while (p > L && k < 64.0f);
    return k;
  }
  // PTRS (Hormann 1993)
  const float b = fmaf(2.53f, __fsqrt_rn(lam), 0.931f);
  const float a = fmaf(0.02483f, b, -0.059f);
  const float inv_alpha = fmaf(1.1328f, __frcp_rn(b - 3.4f), 1.1239f);
  const float vr = 0.9277f - 3.6224f * __frcp_rn(b - 2.0f);
  const float loglam = __logf(lam);
#pragma unroll 1
  for (int it = 0; it < 64; ++it) {
    float u = r.next() - 0.5f;
    float v = r.next();
    float us = 0.5f - fabsf(u);
    float k = floorf(fmaf(fmaf(2.0f * a, __frcp_rn(us), b), u, lam + 0.43f));
    if (us >= 0.07f && v <= vr) return k;
    if (k < 0.0f || (us < 0.013f && v > us)) continue;
    if (__logf(v) * inv_alpha <= fmaf(k, loglam, -lam) - lgammaf(k + 1.0f))
      return k;
  }
  return floorf(lam);
}

// Dims fixed by the reference: [B=1, T=16, C=4, H=540, W=960]
#define DIM_W 960u
#define DIM_H 540u
#define DIM_C 4u
#define DIM_T 16u
#define NROW_ELEMS (DIM_T * DIM_H) /* 8640 floats = 34.5 KB */

__global__ __launch_bounds__(256) void noisy_eld_kernel(
    const f4* __restrict__ img4, const f4* __restrict__ u4,
    const float* __restrict__ nrow, const float* __restrict__ Kp,
    const float* __restrict__ rsigp, const float* __restrict__ tlp,
    const float* __restrict__ rowsigp, const float* __restrict__ cbias,
    const int* __restrict__ ratiop, f4* __restrict__ out4, int nvec) {
  __shared__ __align__(16) float s_nrow[NROW_ELEMS];

  // ---- stage n_row table into LDS once per block ----
#if defined(USE_ASYNC_LDS)
  {
    const int nchunks = NROW_ELEMS / 4;  // 2160 x b128
    float* nr = const_cast<float*>(nrow);
    for (int idx = threadIdx.x; idx < nchunks; idx += blockDim.x) {
      __builtin_amdgcn_global_load_async_to_lds_b128(
          (__attribute__((address_space(1))) v4i*)(nr + 4 * idx),
          (__attribute__((address_space(3))) v4i*)(s_nrow + 4 * idx),
          0, 0);
    }
    __builtin_amdgcn_s_wait_asynccnt(0);
  }
#else
  for (int idx = threadIdx.x; idx < (int)NROW_ELEMS; idx += blockDim.x)
    s_nrow[idx] = nrow[idx];
#endif
  __syncthreads();

  // ---- per-thread uniform constants ----
  const float absK = fabsf(Kp[0]);
  const float invK = __frcp_rn(absK);
  const float rsig = rsigp[0];
  const float tl = tlp[0];
  const float inv_tl = __frcp_rn(tl);
  const float rowsig = rowsigp[0];
  const float factor = (float)ratiop[0] * (1.0f / 15359.0f);  // ratio/SCALE
  const float cb0 = cbias[0], cb1 = cbias[1], cb2 = cbias[2], cb3 = cbias[3];

  const int stride = (int)(gridDim.x * blockDim.x);
  for (int i = (int)(blockIdx.x * blockDim.x + threadIdx.x); i < nvec;
       i += stride) {
    if (i + stride < nvec) {
      __builtin_prefetch(img4 + i + stride, 0, 1);  // global_prefetch_b8
      __builtin_prefetch(u4 + i + stride, 0, 1);
    }
    // touch-once stream: non-temporal b128 loads
    f4 x = __builtin_nontemporal_load(img4 + i);
    f4 u = __builtin_nontemporal_load(u4 + i);

    // decode [t,c,h] for this vec4 (w..w+3 share the row; 960 % 4 == 0)
    const uint32_t e = (uint32_t)i * 4u;
    const uint32_t rw = e / DIM_W;         // (t*4 + c)*540 + h
    const uint32_t tc = rw / DIM_H;        // t*4 + c
    const uint32_t h = rw - tc * DIM_H;
    const uint32_t c = tc & 3u;
    const float rowv = s_nrow[(tc >> 2) * DIM_H + h] * rowsig;
    const float cb = (c & 2u) ? ((c & 1u) ? cb3 : cb2)
                              : ((c & 1u) ? cb1 : cb0);
    const float base = rowv + cb;

    f4 o;
#pragma unroll
    for (int j = 0; j < 4; ++j) {
      const float xv = x[j];
      // img + shot == P * |K|  (Poisson sample times gain)
      const float lam = fmaxf(xv, 0.0f) * invK;
      const float P = poissonf(lam, ((uint64_t)(e + j) + 1ull) * GOLD ^ 0x42ull);
      // Tukey-lambda read noise: (U^l - (1-U)^l)/l * sigma
      const float U = fmaf(u[j], 1.0f - 2e-10f, 1e-10f);
      const float rd = (__powf(U, tl) - __powf(1.0f - U, tl)) * inv_tl * rsig;
      const float acc = fmaf(P, absK, rd + base);
      o[j] = fminf(acc * factor, 1.0f);
    }
    __builtin_nontemporal_store(o, out4 + i);
  }
}

extern "C" void kernel_launch(void* const* d_in, const int* in_sizes, int n_in,
                              void* d_out, int out_size, void* d_ws,
                              size_t ws_size, hipStream_t stream) {
  (void)n_in; (void)d_ws; (void)ws_size; (void)out_size;
  const f4* img4 = (const f4*)d_in[0];
  const float* Kp = (const float*)d_in[1];
  const float* rsig = (const float*)d_in[2];
  const float* tl = (const float*)d_in[3];
  const float* rowsig = (const float*)d_in[4];
  const float* cbias = (const float*)d_in[5];
  const f4* u4 = (const f4*)d_in[6];
  const float* nrow = (const float*)d_in[7];
  const int* ratio = (const int*)d_in[8];
  f4* out4 = (f4*)d_out;

  const int nvec = in_sizes[0] / 4;  // 8,294,400 vec4 elements
  const int threads = 256;           // 8 wave32 waves per block
  int blocks = (nvec + threads - 1) / threads;
  if (blocks > 4096) blocks = 4096;  // grid-stride; amortize LDS staging
  noisy_eld_kernel<<<blocks, threads, 0, stream>>>(
      img4, u4, nrow, Kp, rsig, tl, rowsig, cbias, ratio, out4, nvec);
}